// WTTERNN_32504312496404
// MI455X (gfx1250) — compile-verified
//
#include <hip/hip_runtime.h>
#include <hip/hip_bf16.h>
#include <stdint.h>

// GRU: B=8192 seqs, T=2048 steps, F=1, H=2.  One lane PAIR per sequence
// (lane&1 == hidden unit) -> 512 independent waves, launched as 512
// single-wave workgroups for maximal WGP spread.  All HBM traffic staged
// through LDS with CDNA5 async global<->LDS ops, double-buffered on ASYNCcnt.

#define B_DIM   8192
#define T_DIM   2048
#define TT      32                 // timesteps per tile
#define NTILE   (T_DIM / TT)       // 64
#define ROWS    16                 // batch rows per wave (32 lanes / 2)
#define XSTR    36                 // x-tile row stride (floats): 16B-aligned, conflict-free
#define OSTR    66                 // out-tile row stride (floats): conflict-free
#define PER_WAVE_F (2*ROWS*XSTR + 2*ROWS*OSTR)   // 1152 + 2112 = 3264 floats (~12.75KB)

__device__ __forceinline__ uint32_t lds_off(const void* p) {
  // generic shared address: aperture lives in bits [63:32]; low 32 = LDS byte offset
  return (uint32_t)(uintptr_t)p;
}

__device__ __forceinline__ void async_ld_b32(uint32_t lds, const float* g) {
  asm volatile("global_load_async_to_lds_b32 %0, %1, off"
               :: "v"(lds), "v"(g) : "memory");
}
__device__ __forceinline__ void async_st_b32(float* g, uint32_t lds) {
  asm volatile("global_store_async_from_lds_b32 %0, %1, off"
               :: "v"(g), "v"(lds) : "memory");
}
#define WAIT_ASYNC(N) asm volatile("s_wait_asynccnt %0" :: "n"(N) : "memory")
#define WAIT_DS0()    asm volatile("s_wait_dscnt 0" ::: "memory")

__device__ __forceinline__ float ftanh(float x) {
#if __has_builtin(__builtin_amdgcn_tanhf)
  return __builtin_amdgcn_tanhf(x);
#elif __has_builtin(__builtin_amdgcn_tanh_f32)
  return __builtin_amdgcn_tanh_f32(x);
#else
  float e = __builtin_amdgcn_exp2f(x * 2.8853900817779268f);   // e^{2x}
  return 1.0f - 2.0f * __builtin_amdgcn_rcpf(e + 1.0f);
#endif
}

__global__ __launch_bounds__(32)
void gru_alphabeta_kernel(const float* __restrict__ x,
                          const float* __restrict__ w_ih,
                          const float* __restrict__ w_hh,
                          const float* __restrict__ b_ih,
                          const float* __restrict__ b_hh,
                          float* __restrict__ out)
{
  __shared__ __align__(16) float smem[PER_WAVE_F];

  const int lane = threadIdx.x & 31;
  const int p    = lane >> 1;        // batch row within this wave's 16
  const int u    = lane & 1;         // hidden unit this lane owns
  const size_t b0 = (size_t)blockIdx.x * ROWS;

  float* xs = &smem[0];                   // [2][ROWS][XSTR]
  float* os = &smem[2 * ROWS * XSTR];     // [2][ROWS][OSTR]

  // ---- per-lane constants: this lane handles gates u (r), 2+u (z), 4+u (n).
  // 0.5-scales folded in so sigmoid(x) = 0.5*tanh(pre') + 0.5 with pre' computed
  // directly, and n-gate uses gh_n/2 so sig(pre_r)*gh_n folds to one post-tanh fma.
  const int gr = u, gz = 2 + u, gn = 4 + u;
  const float wr  = 0.5f * w_ih[gr],               wz  = 0.5f * w_ih[gz];
  const float br  = 0.5f * (b_ih[gr] + b_hh[gr]),  bz  = 0.5f * (b_ih[gz] + b_hh[gz]);
  const float ars = 0.5f * w_hh[gr*2 + u],         aro = 0.5f * w_hh[gr*2 + (u^1)];
  const float azs = 0.5f * w_hh[gz*2 + u],         azo = 0.5f * w_hh[gz*2 + (u^1)];
  const float wn  = w_ih[gn],                      bnx = b_ih[gn];
  const float ans = 0.5f * w_hh[gn*2 + u];
  const float ano = 0.5f * w_hh[gn*2 + (u^1)];
  const float bnh = 0.5f * b_hh[gn];

  float h = 0.0f;

  auto prefetch = [&](int tile, int buf) {
#pragma unroll
    for (int r = 0; r < ROWS; ++r) {            // 16 async loads, 128B coalesced each
      const float* gp = x + (b0 + r) * (size_t)T_DIM + (size_t)tile * TT + lane;
      async_ld_b32(lds_off(xs + buf * ROWS * XSTR + r * XSTR + lane), gp);
    }
  };

  auto step = [&](float xv, float* od2) {       // od2 points at orow[2k]
    float ho  = __shfl_xor(h, 1, 32);           // partner's hidden unit (v_permlane)
    float pr  = fmaf(xv, wr, fmaf(h, ars, fmaf(ho, aro, br)));   // pre_r/2
    float pz  = fmaf(xv, wz, fmaf(h, azs, fmaf(ho, azo, bz)));   // pre_z/2
    float nh2 = fmaf(h, ans, fmaf(ho, ano, bnh));                // gh_n/2
    float nx  = fmaf(xv, wn, bnx);
    float thr = ftanh(pr);                      // 2*sig(pre_r)-1
    float nxh = nx + nh2;                       // off critical path (|| tanh)
    float nn  = ftanh(fmaf(thr, nh2, nxh));     // tanh(nx + sig(pre_r)*gh_n)
    float zz  = fmaf(0.5f, ftanh(pz), 0.5f);
    h = fmaf(zz, h - nn, nn);                   // (1-z)*n + z*h
    float tt = ftanh(h);
    // alpha = 2.5*e^tt = exp2(tt*log2e + log2 2.5) ; beta = 4*sig(tt) = 2+2*tanh(tt/2)
    float av = __builtin_amdgcn_exp2f(fmaf(tt, 1.4426950408889634f, 1.3219280948873623f));
    float bv = fmaf(2.0f, ftanh(0.5f * tt), 2.0f);
    od2[u] = u ? bv : av;                       // interleaved (alpha,beta)
  };

  prefetch(0, 0);
  for (int tile = 0; tile < NTILE; ++tile) {
    const int buf = tile & 1;
    if (tile + 1 < NTILE) prefetch(tile + 1, buf ^ 1);

    // <=16 pending: in-order loads => x tile `tile` resident;
    // in-order stores => out-stores from tile-2 drained (out buffer reusable)
    WAIT_ASYNC(16);

    float* xrow = xs + buf * ROWS * XSTR + p * XSTR;
    float* orow = os + buf * ROWS * OSTR + p * OSTR;
#pragma unroll
    for (int k = 0; k < TT; k += 4) {
      float4 xq = *(const float4*)(xrow + k);   // ds_load_b128, pair-broadcast
      step(xq.x, orow + 2 * k + 0);
      step(xq.y, orow + 2 * k + 2);
      step(xq.z, orow + 2 * k + 4);
      step(xq.w, orow + 2 * k + 6);
    }

    WAIT_DS0();                                 // out tile fully in LDS
#pragma unroll
    for (int r = 0; r < ROWS; ++r) {            // 32 async stores, 128B coalesced each
      float* gp = out + ((b0 + r) * (size_t)T_DIM + (size_t)tile * TT) * 2;
      async_st_b32(gp + lane,
                   lds_off(os + buf * ROWS * OSTR + r * OSTR + lane));
      async_st_b32(gp + 32 + lane,
                   lds_off(os + buf * ROWS * OSTR + r * OSTR + 32 + lane));
    }
  }
  WAIT_ASYNC(0);
}

extern "C" void kernel_launch(void* const* d_in, const int* in_sizes, int n_in,
                              void* d_out, int out_size, void* d_ws, size_t ws_size,
                              hipStream_t stream) {
  const float* x    = (const float*)d_in[0];   // (8192, 2048, 1)
  const float* w_ih = (const float*)d_in[1];   // (6, 1)
  const float* w_hh = (const float*)d_in[2];   // (6, 2)
  const float* b_ih = (const float*)d_in[3];   // (6,)
  const float* b_hh = (const float*)d_in[4];   // (6,)
  float* out = (float*)d_out;                  // (8192, 2048, 2)

  const int blocks = B_DIM / ROWS;             // 512 single-wave workgroups
  gru_alphabeta_kernel<<<dim3(blocks), dim3(32), 0, stream>>>(
      x, w_ih, w_hh, b_ih, b_hh, out);
}